// MHI_55662776156695
// MI455X (gfx1250) — compile-verified
//
#include <hip/hip_runtime.h>
#include <math.h>

#define KROWS 100000
#define MDIM  512
#define TILES (KROWS / 16)          // 6250
#define NB3   200                   // blocks for pass 2
#define NW3   (NB3 * 8)             // 1600 waves in pass 2

typedef __attribute__((ext_vector_type(2))) float v2f;
typedef __attribute__((ext_vector_type(8))) float v8f;

// ws layout (float offsets)
#define WS_STATS   0        // [0]=cb, [1]=C, [2]=m, [3]=invZ
#define WS_V       16       // 512
#define WS_V2      528      // 512
#define WS_ZERO    1040     // 512 (zero vector for B lanes n>=2)
#define WS_E       1552     // 100000
#define WS_E2P     101552   // 12500
#define WS_VPART   114052   // 16*512
#define WS_V2PART  122244   // 16*512
#define WS_COLPART 130436   // NB3*512

// ---- kernel 0a: partial v = a1^T W, v2 = a2^T W (16 blocks x 512) ----
__global__ void k0a(const float* __restrict__ W, const float* __restrict__ a,
                    float* __restrict__ vpart, float* __restrict__ v2part) {
  int k = threadIdx.x;
  int bb = blockIdx.x;
  float av = 0.f, av2 = 0.f;
  int j0 = bb * 32;
  for (int j = j0; j < j0 + 32; ++j) {
    float w = W[j * MDIM + k];
    av  += a[j] * w;
    av2 += a[MDIM + j] * w;
  }
  vpart[bb * MDIM + k]  = av;
  v2part[bb * MDIM + k] = av2;
}

// ---- kernel 0b: reduce partials, compute cb = b.(a1+a2), zero zbuf (1 x 512) ----
__global__ void k0b(const float* __restrict__ vpart, const float* __restrict__ v2part,
                    const float* __restrict__ b, const float* __restrict__ a,
                    float* __restrict__ v, float* __restrict__ v2,
                    float* __restrict__ zbuf, float* __restrict__ stats) {
  int k = threadIdx.x;
  float sv = 0.f, sv2 = 0.f;
  for (int bb = 0; bb < 16; ++bb) {
    sv  += vpart[bb * MDIM + k];
    sv2 += v2part[bb * MDIM + k];
  }
  v[k] = sv; v2[k] = sv2; zbuf[k] = 0.f;
  __shared__ float red[512];
  red[k] = b[k] * (a[k] + a[MDIM + k]);
  __syncthreads();
  for (int s = 256; s > 0; s >>= 1) {
    if (k < s) red[k] += red[k + s];
    __syncthreads();
  }
  if (k == 0) stats[0] = red[0];
}

// ---- kernel 1: WMMA dual matvec: e = x.v (stored), e2 = x.v2 (per-tile partial) ----
__global__ void k1_wmma(const float* __restrict__ x, const float* __restrict__ v,
                        const float* __restrict__ v2, const float* __restrict__ zbuf,
                        float* __restrict__ e, float* __restrict__ e2part) {
  int wave = (blockIdx.x * blockDim.x + threadIdx.x) >> 5;
  if (wave >= TILES) return;
  int lane = threadIdx.x & 31;
  int n    = lane & 15;
  int khi  = lane >> 4;
  int r0   = wave * 16;

  // A (16x4 f32): lane supplies row n, K = k + 2*khi + {0,1}
  const float* arow = x + (size_t)(r0 + n) * MDIM + 2 * khi;
  // B (4x16 f32): lane supplies column n of Vmat, K = k + 2*khi + {0,1}
  const float* bsel = (n == 0) ? v : ((n == 1) ? v2 : zbuf);
  const float* brow = bsel + 2 * khi;

  v8f c = {};
  #pragma unroll 8
  for (int k = 0; k < MDIM; k += 4) {
    v2f af = *(const v2f*)(arow + k);
    v2f bf = *(const v2f*)(brow + k);
    c = __builtin_amdgcn_wmma_f32_16x16x4_f32(false, af, false, bf,
                                              (short)0, c, false, false);
  }

  // C layout: VGPR r, lanes 0-15: (M=r, N=lane); lanes 16-31: (M=8+r, N=lane-16)
  if (n == 0) {                       // column 0 = e for rows r0+8*khi .. +7
    #pragma unroll
    for (int r = 0; r < 8; ++r) e[r0 + khi * 8 + r] = c[r];
  } else if (n == 1) {                // column 1 = e2; sum my 8 rows
    float s = c[0] + c[1] + c[2] + c[3] + c[4] + c[5] + c[6] + c[7];
    e2part[wave * 2 + khi] = s;
  }
}

// ---- kernel 2: softmax stats over e (1 x 1024) ----
__global__ void k2_stats(const float* __restrict__ e, const float* __restrict__ e2part,
                         float* __restrict__ stats) {
  __shared__ float red[1024];
  int t = threadIdx.x;
  // E2 = sum of per-tile partials
  float s = 0.f;
  for (int i = t; i < 2 * TILES; i += 1024) s += e2part[i];
  red[t] = s; __syncthreads();
  for (int w = 512; w > 0; w >>= 1) { if (t < w) red[t] += red[t + w]; __syncthreads(); }
  float C = stats[0] + red[0] * (1.0f / (float)KROWS);
  __syncthreads();
  // max over e
  float mx = -INFINITY;
  for (int i = t; i < KROWS; i += 1024) mx = fmaxf(mx, e[i]);
  red[t] = mx; __syncthreads();
  for (int w = 512; w > 0; w >>= 1) { if (t < w) red[t] = fmaxf(red[t], red[t + w]); __syncthreads(); }
  float m = fmaxf(0.0f, red[0] + C);   // relu is monotonic
  __syncthreads();
  // Z = sum exp(relu(e+C) - m)
  float z = 0.f;
  for (int i = t; i < KROWS; i += 1024) z += __expf(fmaxf(0.0f, e[i] + C) - m);
  red[t] = z; __syncthreads();
  for (int w = 512; w > 0; w >>= 1) { if (t < w) red[t] += red[t + w]; __syncthreads(); }
  if (t == 0) { stats[1] = C; stats[2] = m; stats[3] = 1.0f / red[0]; }
}

// ---- kernel 3: pass 2, per-block partial of sum_i alpha_i * x_i (NB3 x 256) ----
__global__ void k3_wsum(const float* __restrict__ x, const float* __restrict__ e,
                        const float* __restrict__ stats, float* __restrict__ colpart) {
  const float C = stats[1], m = stats[2], invZ = stats[3];
  int t = threadIdx.x, lane = t & 31, w = t >> 5;
  int gw = blockIdx.x * 8 + w;   // 0..NW3-1
  float4 acc[4] = {};
  for (int row = gw; row < KROWS; row += NW3) {
    if (row + NW3 < KROWS)
      __builtin_prefetch(x + (size_t)(row + NW3) * MDIM + lane * 16, 0, 0);
    float alpha = __expf(fmaxf(0.0f, e[row] + C) - m) * invZ;
    const float4* xr = (const float4*)(x + (size_t)row * MDIM);
    #pragma unroll
    for (int cc = 0; cc < 4; ++cc) {
      float4 xv = xr[cc * 32 + lane];
      acc[cc].x += alpha * xv.x; acc[cc].y += alpha * xv.y;
      acc[cc].z += alpha * xv.z; acc[cc].w += alpha * xv.w;
    }
  }
  __shared__ float lds[8 * MDIM];
  float* mine = lds + w * MDIM;
  #pragma unroll
  for (int cc = 0; cc < 4; ++cc) {
    int base = cc * 128 + lane * 4;
    mine[base + 0] = acc[cc].x; mine[base + 1] = acc[cc].y;
    mine[base + 2] = acc[cc].z; mine[base + 3] = acc[cc].w;
  }
  __syncthreads();
  for (int col = t; col < MDIM; col += 256) {
    float sum = 0.f;
    for (int ww = 0; ww < 8; ++ww) sum += lds[ww * MDIM + col];
    colpart[blockIdx.x * MDIM + col] = sum;
  }
}

// ---- kernel 4: final reduce of block partials into d_out (1 x 512) ----
__global__ void k4_final(const float* __restrict__ colpart, float* __restrict__ out) {
  int k = threadIdx.x;
  float s = 0.f;
  for (int bb = 0; bb < NB3; ++bb) s += colpart[bb * MDIM + k];
  out[k] = s;
}

extern "C" void kernel_launch(void* const* d_in, const int* in_sizes, int n_in,
                              void* d_out, int out_size, void* d_ws, size_t ws_size,
                              hipStream_t stream) {
  const float* x = (const float*)d_in[0];   // (100000, 512)
  const float* W = (const float*)d_in[1];   // (512, 512)
  const float* b = (const float*)d_in[2];   // (512,)
  const float* a = (const float*)d_in[3];   // (1024, 1)
  float* out = (float*)d_out;               // (512,)
  float* ws  = (float*)d_ws;

  float* stats   = ws + WS_STATS;
  float* v       = ws + WS_V;
  float* v2      = ws + WS_V2;
  float* zbuf    = ws + WS_ZERO;
  float* e       = ws + WS_E;
  float* e2part  = ws + WS_E2P;
  float* vpart   = ws + WS_VPART;
  float* v2part  = ws + WS_V2PART;
  float* colpart = ws + WS_COLPART;

  k0a<<<16, 512, 0, stream>>>(W, a, vpart, v2part);
  k0b<<<1, 512, 0, stream>>>(vpart, v2part, b, a, v, v2, zbuf, stats);
  int blocks1 = (TILES * 32 + 255) / 256;   // 782
  k1_wmma<<<blocks1, 256, 0, stream>>>(x, v, v2, zbuf, e, e2part);
  k2_stats<<<1, 1024, 0, stream>>>(e, e2part, stats);
  k3_wsum<<<NB3, 256, 0, stream>>>(x, e, stats, colpart);
  k4_final<<<1, 512, 0, stream>>>(colpart, out);
}